// EpsiLayer_62122406969385
// MI455X (gfx1250) — compile-verified
//
#include <hip/hip_runtime.h>

// out[b,t,j] = P[b,t,j] + sum_{s=0..t} g[s,j] * P[b,t-s,j]
// Per channel j: y = L p, L lower-triangular Toeplitz, L[t,s] = g[t-s,j].
// Tiled 16x16 triangular block-GEMM on V_WMMA_F32_16X16X4_F32.
// Wave = 4 consecutive output tiles; lag loop with shared Toeplitz A, 4 accs,
// register-resident sliding window of 4 B tiles. Hot loop is branch-free; the
// triangular tail (d = tb..tb+3) is a fully unrolled epilogue.

typedef __attribute__((ext_vector_type(2))) float v2f;
typedef __attribute__((ext_vector_type(8))) float v8f;

namespace {
constexpr int B_  = 8;
constexpr int T_  = 4096;
constexpr int NR_ = 256;
constexpr int NT  = T_ / 16;       // 256 time tiles
constexpr int NG  = NT / 4;        // 64 groups of 4 output tiles
constexpr int PROW = T_ + 4;       // padded LDS row stride (floats)
constexpr int GPAD = 16;           // zero front-pad for g (no d==0 guards)
constexpr int NTHREADS = 256;
constexpr int NWAVE = NTHREADS / 32;
}

// K-relabeling within a slice: both A and B pairs are stored "reversed" so the
// A pair comes straight out of an ascending ds_load_2addr into operand order.
#define LOADA(D)                                                              \
  {                                                                           \
    const int gb = 16 * (D) + M - kb;                                         \
    _Pragma("unroll")                                                         \
    for (int q = 0; q < 4; ++q) {                                             \
      a[q].x = gp[gb - 4 * q - 1];                                            \
      a[q].y = gp[gb - 4 * q];                                                \
    }                                                                         \
  }

#define LOADB(SLOT, SI)                                                       \
  {                                                                           \
    const float* pr = prow0 + 16 * (SI);                                      \
    _Pragma("unroll")                                                         \
    for (int q = 0; q < 4; ++q) {                                             \
      Bt[(SLOT) & 3][q].x = pr[4 * q + 1];                                    \
      Bt[(SLOT) & 3][q].y = pr[4 * q];                                        \
    }                                                                         \
  }

#define WMMA4(ACC, SLOT)                                                      \
  {                                                                           \
    _Pragma("unroll")                                                         \
    for (int q = 0; q < 4; ++q)                                               \
      ACC = __builtin_amdgcn_wmma_f32_16x16x4_f32(                            \
          false, a[q], false, Bt[(SLOT) & 3][q], (short)0, ACC, false, false);\
  }

#define STORE_TILE(ACC, C)                                                    \
  {                                                                           \
    _Pragma("unroll")                                                         \
    for (int r = 0; r < 8; ++r) {                                             \
      const int t = 16 * (tb + (C)) + mbase + r;                              \
      out[((size_t)N * T_ + t) * NR_ + j] = ACC[r] + prN[t];                  \
    }                                                                         \
  }

__global__ __launch_bounds__(NTHREADS)
void epsi_causal_toeplitz_wmma(const float* __restrict__ P,
                               const float* __restrict__ g,
                               float* __restrict__ out) {
  // 8*PROW + GPAD + T floats = 147,648 bytes (of 320 KB WGP LDS)
  __shared__ float smem[B_ * PROW + GPAD + T_];
  float* Plds = smem;                   // [b][t], row stride PROW
  float* gpad = smem + B_ * PROW;       // [GPAD zeros | g[:,j]]

  const int j   = blockIdx.x;           // channel
  const int tid = threadIdx.x;

  // ---- stage P[:, :, j] (strided gather; P fits in 192MB L2) and g[:, j] ----
  for (int idx = tid; idx < B_ * T_; idx += NTHREADS) {
    const int b = idx >> 12;            // idx = b*T + t
    const int t = idx & (T_ - 1);
    Plds[b * PROW + t] = P[(size_t)idx * NR_ + j];
  }
  if (tid < GPAD) gpad[tid] = 0.0f;
  for (int t = tid; t < T_; t += NTHREADS) {
    gpad[GPAD + t] = g[(size_t)t * NR_ + j];
  }
  __syncthreads();

  const int lane = tid & 31;
  // readfirstlane: wave id is physically uniform; make it scalar so all
  // tb-derived control flow is SALU (EXEC stays all-ones around WMMA).
  const int wave = __builtin_amdgcn_readfirstlane(tid >> 5);
  const int M    = lane & 15;             // A row (output time within tile)
  const int kb   = (lane < 16) ? 0 : 2;   // f32 WMMA K base for this lane half
  const int bcol = lane & 7;              // batch column; lanes 8-15 duplicate
  const float* gp    = gpad + GPAD;       // gp[i] valid for i >= -GPAD
  const float* prow0 = Plds + bcol * PROW + kb;

  for (int gi = wave; gi < NG; gi += NWAVE) {
    const int tb = 4 * gi;                // tb % 4 == 0 (slot math constant-folds)
    v8f acc0 = {}, acc1 = {}, acc2 = {}, acc3 = {};
    v2f Bt[4][4];                         // sliding window of B tiles [slot][q]
    v2f a[4];                             // shared Toeplitz A slices for lag d

    // prime window with s-tiles tb+1..tb+3 (slot = si & 3 = c)
    #pragma unroll
    for (int c = 1; c < 4; ++c) LOADB(c, tb + c);

    // ---- branch-free main lag loop: d = 0 .. tb-1 ----
    for (int d4 = 0; d4 < tb; d4 += 4) {
      #pragma unroll
      for (int p = 0; p < 4; ++p) {
        const int d = d4 + p;
        LOADB(0 - p, tb - d);             // slide window: si = tb-d >= 1 here
        LOADA(d);
        WMMA4(acc0, 0 - p);
        WMMA4(acc1, 1 - p);
        WMMA4(acc2, 2 - p);
        WMMA4(acc3, 3 - p);
      }
    }

    // ---- triangular epilogue: d = tb .. tb+3 ----
    LOADB(0, 0);                          // si = 0 into slot 0
    LOADA(tb);
    WMMA4(acc0, 0); WMMA4(acc1, 1); WMMA4(acc2, 2); WMMA4(acc3, 3);
    LOADA(tb + 1);
    WMMA4(acc1, 0); WMMA4(acc2, 1); WMMA4(acc3, 2);
    LOADA(tb + 2);
    WMMA4(acc2, 0); WMMA4(acc3, 1);
    LOADA(tb + 3);
    WMMA4(acc3, 0);

    // ---- store: D layout = VGPR r: lanes 0-15 -> M=r, lanes 16-31 -> M=8+r
    const int N     = lane & 15;
    const int mbase = (lane < 16) ? 0 : 8;
    if (N < B_) {
      const float* prN = Plds + N * PROW;
      STORE_TILE(acc0, 0)
      STORE_TILE(acc1, 1)
      STORE_TILE(acc2, 2)
      STORE_TILE(acc3, 3)
    }
  }
}

extern "C" void kernel_launch(void* const* d_in, const int* in_sizes, int n_in,
                              void* d_out, int out_size, void* d_ws, size_t ws_size,
                              hipStream_t stream) {
  const float* P = (const float*)d_in[0];   // (B, T, NR) fp32
  const float* g = (const float*)d_in[1];   // (T, NR)   fp32
  float* out = (float*)d_out;               // (B, T, NR) fp32
  (void)in_sizes; (void)n_in; (void)out_size; (void)d_ws; (void)ws_size;

  epsi_causal_toeplitz_wmma<<<dim3(NR_), dim3(NTHREADS), 0, stream>>>(P, g, out);
}